// SelfAttention_695784702010
// MI455X (gfx1250) — compile-verified
//
#include <hip/hip_runtime.h>
#include <stdint.h>

// ---------------------------------------------------------------------------
// Fused head-axis attention for MI455X (gfx1250, wave32).
//   out = ( softmax_over_heads( (vWq)(vWk)^T / sqrt(32) ) @ (vWv) ) @ Wp + bp
// Four 65536x256x256 fp32 GEMMs on V_WMMA_F32_16X16X4_F32 with each weight
// B-fragment reused across two stacked M-tiles (16 FLOP/L2-byte). Input tiles
// are DMA'd by the Tensor Data Mover into a double-buffered LDS pad with
// s_wait_tensorcnt pipelining; the attention x-tile aliases the dead input
// buffer to keep LDS at 160 KB (2 workgroups / WGP).
// ---------------------------------------------------------------------------

#define DIM          256
#define NHEADS       8
#define HD           32
#define TILE_M       32
#define TILE_ELTS    (TILE_M * DIM)      // 8192 f32 = 32 KB, contiguous
#define TILES_PER_WG 4

typedef float v2f __attribute__((ext_vector_type(2)));
typedef float v8f __attribute__((ext_vector_type(8)));
typedef int   i32x4 __attribute__((ext_vector_type(4)));
typedef int   i32x8 __attribute__((ext_vector_type(8)));

// ---- Tensor Data Mover: 1-D 32 KB tile  global -> LDS  (D# in SGPRs) -------
// Group0: [1:0]=count=1 | [63:32]=lds_addr | [120:64]=global_addr | [127:126]=2
// Group1: data_size=4B, tensor_dim0=stride=8192, tile_dim0=8192 (1-D tile)
static __device__ __forceinline__ void tdm_load_tile(const float* gsrc,
                                                     unsigned lds_off)
{
    const unsigned long long ga = (unsigned long long)(uintptr_t)gsrc;
    i32x4 g0;
    g0[0] = 1;                                              // count=1 (valid)
    g0[1] = (int)lds_off;                                   // LDS byte address
    g0[2] = (int)(unsigned)(ga & 0xffffffffull);            // global addr lo
    g0[3] = (int)((((unsigned)(ga >> 32)) & 0x01ffffffu)    // addr[56:32]
                  | (2u << 30));                            // type = 2 (image)
    i32x8 g1;
    g1[0] = 0x00020000;                  // workgroup_mask=0, data_size=2 (4B)
    g1[1] = (int)(TILE_ELTS << 16);      // tensor_dim0[15:0]  -> bits[63:48]
    g1[2] = (int)(1u << 16);             // tensor_dim0[31:16]=0, tensor_dim1=1
    g1[3] = (int)(TILE_ELTS << 16);      // tile_dim0 -> bits[127:112]
    g1[4] = 0;                           // tile_dim1=0 (1-D), tile_dim2=0
    g1[5] = TILE_ELTS;                   // tensor_dim0_stride lo32
    g1[6] = 0;                           // stride hi16 / dim1_stride lo16
    g1[7] = 0;                           // dim1_stride hi32
    asm volatile("tensor_load_to_lds %0, %1" :: "s"(g0), "s"(g1) : "memory");
}

// Two stacked 16(M) x 16(N) fp32 WMMA tiles of A(32xDIM, LDS) @ W(DIMxDIM,
// global), sharing one B-fragment load per K-step (B reuse across M).
// f32 WMMA operand layout (ISA 7.12.2):
//   lane = 16*lh + r ; A frag = A[row r (or 16+r)][2*lh + {0,1}]
//   B frag = W[2*lh + {0,1}][n0+r] ; C frag j = C[j + 8*lh][n0 + r]
static __device__ __forceinline__ void gemm_rowtile2_f32(
    const float* __restrict__ A, const float* __restrict__ W,
    int n0, int lh, int r, float cinit, v8f& acc0, v8f& acc1)
{
    v8f a0i = {cinit, cinit, cinit, cinit, cinit, cinit, cinit, cinit};
    acc0 = a0i;
    acc1 = a0i;
    const float* arow0 = A + r * DIM;
    const float* arow1 = A + (r + 16) * DIM;
    #pragma unroll 8
    for (int kk = 0; kk < DIM; kk += 4) {
        const int ka = kk + 2 * lh;                 // even -> 8B-aligned
        v2f a0 = *(const v2f*)(arow0 + ka);         // ds_load_b64
        v2f a1 = *(const v2f*)(arow1 + ka);
        const float* wp = W + ka * DIM + n0 + r;    // L2-resident weights
        v2f b;
        b.x = wp[0];
        b.y = wp[DIM];
        acc0 = __builtin_amdgcn_wmma_f32_16x16x4_f32(
            false, a0, false, b, (short)0, acc0, false, false);
        acc1 = __builtin_amdgcn_wmma_f32_16x16x4_f32(
            false, a1, false, b, (short)0, acc1, false, false);
    }
}

static __device__ __forceinline__ void store_tile2_lds(
    float* __restrict__ dst, v8f acc0, v8f acc1, int n0, int lh, int r)
{
    #pragma unroll
    for (int j = 0; j < 8; ++j)
        dst[(j + 8 * lh) * DIM + n0 + r] = acc0[j];
    #pragma unroll
    for (int j = 0; j < 8; ++j)
        dst[(16 + j + 8 * lh) * DIM + n0 + r] = acc1[j];
}

__global__ __launch_bounds__(256, 1) void fused_headattn_kernel(
    const float* __restrict__ v,
    const float* __restrict__ Wq,
    const float* __restrict__ Wk,
    const float* __restrict__ Wv,
    const float* __restrict__ Wp,
    const float* __restrict__ bp,
    float* __restrict__ out)
{
    // [vs0 | vs1 | qs | ks | vvs] = 5 x 32 KB = 160 KB. The x-tile aliases the
    // current input buffer (dead after stage 1; next TDM write to it happens
    // only after the stage-3 barrier of this iteration).
    __shared__ float smem[5 * TILE_ELTS];
    float* const vsbuf0 = smem;
    float* const vsbuf1 = smem + TILE_ELTS;
    float* const qs     = smem + 2 * TILE_ELTS;
    float* const ks     = smem + 3 * TILE_ELTS;
    float* const vvs    = smem + 4 * TILE_ELTS;

    const int t    = threadIdx.x;
    const int wave = t >> 5;          // 8 waves (wave32)
    const int lane = t & 31;
    const int lh   = lane >> 4;       // lane-half selects K pair / M+8
    const int r    = lane & 15;
    const long long m_base = (long long)blockIdx.x * (TILE_M * TILES_PER_WG);

    // Prologue: TDM fetch of tile 0 (EXEC-independent; wave 0 only so
    // TENSORcnt bookkeeping stays in one wave).
    if (wave == 0)
        tdm_load_tile(v + (size_t)m_base * DIM, (unsigned)(uintptr_t)vsbuf0);

    for (int it = 0; it < TILES_PER_WG; ++it) {
        float* const cvs = (it & 1) ? vsbuf1 : vsbuf0;   // current input tile
        float* const nvs = (it & 1) ? vsbuf0 : vsbuf1;   // prefetch target
        float* const xs  = cvs;                          // alias (see above)

        if (wave == 0) {
            if (it + 1 < TILES_PER_WG) {    // prefetch next tile into other buf
                tdm_load_tile(v + (size_t)(m_base + (it + 1) * TILE_M) * DIM,
                              (unsigned)(uintptr_t)nvs);
                // In-order TDM: <=1 outstanding == tile `it` has landed while
                // tile `it+1` streams in the background.
                __builtin_amdgcn_s_wait_tensorcnt((short)1);
            } else {
                __builtin_amdgcn_s_wait_tensorcnt((short)0);
            }
        }
        __syncthreads();

        const long long m0 = m_base + (long long)it * TILE_M;

        // ---- Stage 1: q/k/vv projections, B reused across stacked M-tiles --
        for (int nt = wave; nt < 16; nt += 8) {
            const int n0 = nt * 16;
            v8f c0, c1;
            gemm_rowtile2_f32(cvs, Wq, n0, lh, r, 0.f, c0, c1);
            store_tile2_lds(qs, c0, c1, n0, lh, r);
            gemm_rowtile2_f32(cvs, Wk, n0, lh, r, 0.f, c0, c1);
            store_tile2_lds(ks, c0, c1, n0, lh, r);
            gemm_rowtile2_f32(cvs, Wv, n0, lh, r, 0.f, c0, c1);
            store_tile2_lds(vvs, c0, c1, n0, lh, r);
        }
        __syncthreads();

        // ---- Stage 2: per-(row, head) 8x8 attention over the HEADS axis ----
        {                                         // 256 workers: 32 rows x 8 heads
            const int row = t >> 3;
            const int h   = t & 7;
            const float scale = 0.17677669529663687f;   // 32^-0.5
            const float* qrow = qs + row * DIM + h * HD;

            float p[NHEADS];
            float mx = -1e30f;
            #pragma unroll
            for (int g = 0; g < NHEADS; ++g) {
                const float* krow = ks + row * DIM + g * HD;
                float s = 0.f;
                #pragma unroll
                for (int d = 0; d < HD; ++d) s += qrow[d] * krow[d];
                s *= scale;
                p[g] = s;
                mx = fmaxf(mx, s);
            }
            float sum = 0.f;
            #pragma unroll
            for (int g = 0; g < NHEADS; ++g) { p[g] = __expf(p[g] - mx); sum += p[g]; }
            const float inv = 1.f / sum;

            float* xrow = xs + row * DIM + h * HD;
            #pragma unroll
            for (int n = 0; n < HD; ++n) {
                float acc = 0.f;
                #pragma unroll
                for (int g = 0; g < NHEADS; ++g)
                    acc += p[g] * vvs[row * DIM + g * HD + n];
                xrow[n] = acc * inv;
            }
        }
        __syncthreads();

        // ---- Stage 3: output projection; bias folded into C init -----------
        for (int nt = wave; nt < 16; nt += 8) {
            const int n0 = nt * 16;
            v8f c0, c1;
            gemm_rowtile2_f32(xs, Wp, n0, lh, r, bp[n0 + r], c0, c1);
            #pragma unroll
            for (int j = 0; j < 8; ++j)
                out[(size_t)(m0 + j + 8 * lh) * DIM + n0 + r] = c0[j];
            #pragma unroll
            for (int j = 0; j < 8; ++j)
                out[(size_t)(m0 + 16 + j + 8 * lh) * DIM + n0 + r] = c1[j];
        }
        __syncthreads();   // qs/ks/vvs and both vs buffers reusable next iter
    }
}

extern "C" void kernel_launch(void* const* d_in, const int* in_sizes, int n_in,
                              void* d_out, int out_size, void* d_ws, size_t ws_size,
                              hipStream_t stream)
{
    const float* v  = (const float*)d_in[0];
    const float* Wq = (const float*)d_in[1];
    const float* Wk = (const float*)d_in[2];
    const float* Wv = (const float*)d_in[3];
    const float* Wp = (const float*)d_in[4];
    const float* bp = (const float*)d_in[5];
    float* out = (float*)d_out;

    const int rows   = in_sizes[0] / DIM;                  // B*N*T = 65536
    const int blocks = rows / (TILE_M * TILES_PER_WG);     // 512 workgroups

    fused_headattn_kernel<<<blocks, 256, 0, stream>>>(v, Wq, Wk, Wv, Wp, bp, out);
}